// ShapeIndex_22711787061653
// MI455X (gfx1250) — compile-verified
//
#include <hip/hip_runtime.h>
#include <math.h>

// Problem constants from the reference (B,C,H,W) = (16,1,1536,1536)
#define Bn 16
#define Hn 1536
#define Wn 1536
#define TW 512     // W-elements per block tile (1536 / 512 = 3 tiles per row)
#define TPB 128    // threads per block; each thread produces 4 outputs (float4)

typedef __attribute__((ext_vector_type(4))) float v4f;

__device__ __forceinline__ float rmnan(float v) { return v == v ? v : 0.0f; }
__device__ __forceinline__ float frcp(float v)  { return __builtin_amdgcn_rcpf(v); }

__global__ __launch_bounds__(TPB) void si_kernel(const float* __restrict__ x,
                                                 float* __restrict__ out) {
    // Row layout: [0..1] pad, [2..3] halo (w0-2, w0-1), [4..515] tile data.
    // Stride 520 floats keeps every row base 16B-aligned (520*4 = 2080 = 130*16).
    __shared__ float lds[3][520];

    const int tid = threadIdx.x;
    // 3-D grid: x = W-tile (0..2), y = h (0..1535), z = b (0..15).
    const int tile = blockIdx.x;
    const int h    = blockIdx.y;
    const int b    = blockIdx.z;
    const int w0   = tile * TW;

    // Circular wrap via selects (no integer division).
    const int hm1 = (h == 0) ? (Hn - 1) : (h - 1);
    const int hm2 = (h <= 1) ? (h + Hn - 2) : (h - 2);
    const int rows[3] = { h, hm1, hm2 };

    const size_t plane = (size_t)b * Hn;

    // ---- Async global -> LDS staging (CDNA5 ASYNCcnt path) ----
    // Bulk: 512 floats per row = 128 lanes x 16B (16B-aligned both sides).
    #pragma unroll
    for (int r = 0; r < 3; ++r) {
        const float* g = x + (plane + rows[r]) * Wn + w0 + 4 * tid;
        unsigned l = (unsigned)(size_t)&lds[r][4 + 4 * tid];
        asm volatile("global_load_async_to_lds_b128 %0, %1, off"
                     :: "v"(l), "v"((unsigned long long)(size_t)g)
                     : "memory");
    }
    // Halo: two floats at (row, (w0 + W - 2) % W); contiguous since w0 is a
    // multiple of 512 (wrap case w0==0 -> indices W-2, W-1, also contiguous).
    if (tid < 3) {
        const int hw = (w0 == 0) ? (Wn - 2) : (w0 - 2);
        const float* g = x + (plane + rows[tid]) * Wn + hw;
        unsigned l = (unsigned)(size_t)&lds[tid][2];           // 8B-aligned
        asm volatile("global_load_async_to_lds_b64 %0, %1, off"
                     :: "v"(l), "v"((unsigned long long)(size_t)g)
                     : "memory");
    }
    asm volatile("s_wait_asynccnt 0" ::: "memory");
    __syncthreads();

    // ---- Compute 4 outputs per thread from LDS ----
    const int base = 4 * tid;          // lw = base + 4 + j, j = 0..3
    float a0[6], a1[5], a2[4];
    #pragma unroll
    for (int k = 0; k < 6; ++k) a0[k] = lds[0][base + 2 + k]; // x[h][lw-2 .. lw+3]
    #pragma unroll
    for (int k = 0; k < 5; ++k) a1[k] = lds[1][base + 3 + k]; // x[h-1][lw-1 .. lw+3]
    #pragma unroll
    for (int k = 0; k < 4; ++k) a2[k] = lds[2][base + 4 + k]; // x[h-2][lw .. lw+3]

    float res[4];
    #pragma unroll
    for (int j = 0; j < 4; ++j) {
        const float c    = a0[j + 2];   // x[h][w]
        const float w1   = a0[j + 1];   // x[h][w-1]
        const float w2   = a0[j + 0];   // x[h][w-2]
        const float n1   = a1[j + 1];   // x[h-1][w]
        const float n1w1 = a1[j + 0];   // x[h-1][w-1]
        const float n2   = a2[j];       // x[h-2][w]

        // roll(m,1,ax) - m  ==>  m[i-1] - m[i]
        const float p = n1 - c;
        const float q = w1 - c;
        const float r = n2 - 2.0f * n1 + c;
        const float s = (n1w1 - w1) - (n1 - c);
        const float t = w2 - 2.0f * w1 + c;

        const float one_pq = 1.0f + p * p + q * q;
        const float num = (1.0f + q * q) * r - 2.0f * p * q * s + (1.0f + p * p) * t;

        // 1/(2*sqrt(one_pq^3)) == 0.5 * rsqrt(one_pq^3); one_pq >= 1 (safe)
        const float Hc = rmnan(-num * 0.5f *
                               __builtin_amdgcn_rsqf(one_pq * one_pq * one_pq));
        const float Kc = rmnan((r * t - s * s) * frcp(one_pq * one_pq));

        const float root = __builtin_amdgcn_sqrtf(Hc * Hc - Kc); // NaN if negative
        const float kmin = rmnan(Hc - root);
        const float kmax = rmnan(Hc + root);

        // rcp(0)=inf -> atan(+-inf)=+-pi/2; 0*inf=NaN -> rmnan -> 0 (matches ref)
        const float si = rmnan(0.636619772367581343f * // 2/pi
                               atanf((kmax + kmin) * frcp(kmax - kmin)));
        res[j] = si;
    }

    // NT store: don't let the 151MB output stream evict the input from L2.
    v4f o;
    o.x = res[0]; o.y = res[1]; o.z = res[2]; o.w = res[3];
    v4f* dst = (v4f*)(out + (plane + h) * Wn + w0 + base);
    __builtin_nontemporal_store(o, dst);
}

extern "C" void kernel_launch(void* const* d_in, const int* in_sizes, int n_in,
                              void* d_out, int out_size, void* d_ws, size_t ws_size,
                              hipStream_t stream) {
    const float* x = (const float*)d_in[0];
    float* out = (float*)d_out;
    (void)in_sizes; (void)n_in; (void)out_size; (void)d_ws; (void)ws_size;

    dim3 grid(Wn / TW, Hn, Bn);   // (3, 1536, 16)
    si_kernel<<<grid, dim3(TPB), 0, stream>>>(x, out);
}